// TransformLayer_22754736734798
// MI455X (gfx1250) — compile-verified
//
#include <hip/hip_runtime.h>

typedef float float4v __attribute__((ext_vector_type(4)));

// Fast path: C == 128 (32 float4 chunks per row == one wave32).
// Each lane permanently owns 4 channels; reciprocals live in registers.
// Streaming is done with B128 non-temporal loads/stores + global_prefetch.
__global__ __launch_bounds__(256) void transform_scale_c128(
    const float* __restrict__ x,
    const float* __restrict__ max_vals,
    float* __restrict__ out,
    long long nchunks /* total float4 chunks = N*C/4 */)
{
    // chunk-in-row for this lane: blockDim (256) and grid stride are multiples
    // of 32, so (global chunk index) & 31 == threadIdx.x & 31 for every
    // iteration -> channel ownership is loop-invariant.
    const int lane_chunk = threadIdx.x & 31;

    // Per-lane channel reciprocals with zero-guard (computed once).
    const float4v m = *(const float4v*)(max_vals + lane_chunk * 4);
    float4v r;
    r.x = (m.x == 0.0f) ? 0.0f : 1.0f / m.x;
    r.y = (m.y == 0.0f) ? 0.0f : 1.0f / m.y;
    r.z = (m.z == 0.0f) ? 0.0f : 1.0f / m.z;
    r.w = (m.w == 0.0f) ? 0.0f : 1.0f / m.w;

    const float4v* __restrict__ xv = (const float4v*)x;
    float4v* __restrict__ ov = (float4v*)out;

    long long i = (long long)blockIdx.x * blockDim.x + threadIdx.x;
    const long long stride = (long long)gridDim.x * blockDim.x;

    for (; i < nchunks; i += stride) {
        // gfx1250 prefetch path: pull the next grid-stride tile (~32MB) toward
        // L2 while this tile streams through the VALU.
        if (i + stride < nchunks)
            __builtin_prefetch(&xv[i + stride], 0, 0);

        // Streamed once, never re-read -> non-temporal B128 load.
        float4v v = __builtin_nontemporal_load(&xv[i]);
        v.x *= r.x;
        v.y *= r.y;
        v.z *= r.z;
        v.w *= r.w;
        // Written once, never read by us -> non-temporal B128 store.
        __builtin_nontemporal_store(v, &ov[i]);
    }
}

// Generic fallback for any C (scalar, still NT-streamed). Not expected to run
// for this problem (C is fixed at 128), but keeps kernel_launch total.
__global__ __launch_bounds__(256) void transform_scale_generic(
    const float* __restrict__ x,
    const float* __restrict__ max_vals,
    float* __restrict__ out,
    long long n, int C)
{
    long long i = (long long)blockIdx.x * blockDim.x + threadIdx.x;
    const long long stride = (long long)gridDim.x * blockDim.x;
    for (; i < n; i += stride) {
        const int j = (int)(i % C);
        const float m = max_vals[j];
        const float v = __builtin_nontemporal_load(&x[i]);
        const float o = (m == 0.0f) ? 0.0f : v / m;
        __builtin_nontemporal_store(o, &out[i]);
    }
}

extern "C" void kernel_launch(void* const* d_in, const int* in_sizes, int n_in,
                              void* d_out, int out_size, void* d_ws, size_t ws_size,
                              hipStream_t stream) {
    const float* x        = (const float*)d_in[0];
    const float* max_vals = (const float*)d_in[1];
    float* out            = (float*)d_out;

    const long long total = (long long)in_sizes[0];  // N*C elements
    const int C           = in_sizes[1];             // 128

    if (C == 128 && (total & 3LL) == 0LL) {
        const long long nchunks = total >> 2;        // float4 chunks
        long long want = (nchunks + 255) / 256;
        const int blocks = (int)(want > 8192 ? 8192 : (want < 1 ? 1 : want));
        transform_scale_c128<<<blocks, 256, 0, stream>>>(x, max_vals, out, nchunks);
    } else {
        long long want = (total + 255) / 256;
        const int blocks = (int)(want > 16384 ? 16384 : (want < 1 ? 1 : want));
        transform_scale_generic<<<blocks, 256, 0, stream>>>(x, max_vals, out, total, C);
    }
}